// GraphQNetwork_84585085928025
// MI455X (gfx1250) — compile-verified
//
#include <hip/hip_runtime.h>
#include <math.h>

// ---------------------------------------------------------------------------
// GraphQNetwork for MI455X (gfx1250, wave32).
// Edge phases are L2-atomic scatter/gather (memory-bound, ~2 GB traffic ->
// <100us at 23.3 TB/s; per-node tensors live in 192MB L2). Dense node-feature
// transforms use V_WMMA_F32_16X16X4_F32 with [Wl|Wr] packed into one B tile.
// ---------------------------------------------------------------------------

constexpr int B_GRAPHS = 256;
constexpr int NG       = 512;
constexpr int NN       = B_GRAPHS * NG;   // 131072 nodes
constexpr int NE       = NN * 16;         // 2097152 edges
constexpr int ET       = NE + NN;         // edges + self loops
constexpr int FE       = 8;               // edge feature dim

typedef __attribute__((ext_vector_type(2))) float v2f;
typedef __attribute__((ext_vector_type(8))) float v8f;

// -------------------- helpers --------------------

__device__ __forceinline__ void atomic_max_f32(float* addr, float v) {
  // order-preserving int encoding trick; buffer initialized to -inf
  if (v >= 0.0f) atomicMax((int*)addr, __float_as_int(v));
  else           atomicMin((unsigned int*)addr, __float_as_uint(v));
}

__global__ void k_fill(float* __restrict__ p, float v, int n) {
  int i = blockIdx.x * blockDim.x + threadIdx.x;
  if (i < n) p[i] = v;
}

// -------------------- self-loop attr (shared by all layers) --------------------

__global__ void k_loop_scatter(const int* __restrict__ dst,
                               const float* __restrict__ edge_attr,
                               float* __restrict__ deg, float* __restrict__ loop_sum) {
  int e = blockIdx.x * blockDim.x + threadIdx.x;
  if (e >= NE) return;
  int d = dst[e];
  atomicAdd(&deg[d], 1.0f);
  const float* ea = edge_attr + (size_t)e * FE;
#pragma unroll
  for (int j = 0; j < FE; ++j) atomicAdd(&loop_sum[(size_t)d * FE + j], ea[j]);
}

__global__ void k_loop_norm(float* __restrict__ loop_attr, const float* __restrict__ deg) {
  int n = blockIdx.x * blockDim.x + threadIdx.x;
  if (n >= NN) return;
  float inv = 1.0f / fmaxf(deg[n], 1.0f);
#pragma unroll
  for (int j = 0; j < FE; ++j) loop_attr[(size_t)n * FE + j] *= inv;
}

// -------------------- WMMA node transform: xl = h@Wl+bl, xr = h@Wr+br ------
// One wave per 16-node tile. B operand packs Wl in cols [0,CO) and Wr in cols
// [8,8+CO). f32 16x16x4 striping (ISA 7.12.2): A vgpr v, lane L ->
// A[M = L%16][K = (L/16)*2 + v]; same index pattern for B rows; D vgpr r,
// lane L -> D[M = r + (L/16)*8][N = L%16].

template <int CI, int CO>
__global__ void k_node_transform(const float* __restrict__ h,
                                 const float* __restrict__ Wl, const float* __restrict__ bl,
                                 const float* __restrict__ Wr, const float* __restrict__ br,
                                 float* __restrict__ xl, float* __restrict__ xr) {
  const int lane = threadIdx.x & 31;
  const int wave = blockIdx.x * (blockDim.x >> 5) + (threadIdx.x >> 5);
  const int row0 = wave * 16;
  if (row0 >= NN) return;            // wave-uniform: EXEC stays all-1s
  const int half = lane >> 4;
  const int l16  = lane & 15;
  constexpr int KSTEPS = (CI + 3) / 4;
  v8f acc = {};
#pragma unroll
  for (int kk = 0; kk < KSTEPS; ++kk) {
    const int kb = kk * 4 + half * 2;
    const int m  = row0 + l16;
    v2f a, b;
    a.x = (kb + 0 < CI) ? h[(size_t)m * CI + kb + 0] : 0.0f;
    a.y = (kb + 1 < CI) ? h[(size_t)m * CI + kb + 1] : 0.0f;
    float b0 = 0.0f, b1 = 0.0f;
    if (kb + 0 < CI) {
      if (l16 < CO)                       b0 = Wl[(kb + 0) * CO + l16];
      else if (l16 >= 8 && l16 < 8 + CO)  b0 = Wr[(kb + 0) * CO + (l16 - 8)];
    }
    if (kb + 1 < CI) {
      if (l16 < CO)                       b1 = Wl[(kb + 1) * CO + l16];
      else if (l16 >= 8 && l16 < 8 + CO)  b1 = Wr[(kb + 1) * CO + (l16 - 8)];
    }
    b.x = b0; b.y = b1;
    acc = __builtin_amdgcn_wmma_f32_16x16x4_f32(
        /*neg_a=*/false, a, /*neg_b=*/false, b,
        /*c_mod=*/(short)0, acc, /*reuse_a=*/false, /*reuse_b=*/false);
  }
#pragma unroll
  for (int r = 0; r < 8; ++r) {
    const int   m = row0 + r + half * 8;
    const float v = acc[r];
    if (l16 < CO)                      xl[(size_t)m * CO + l16]       = v + bl[l16];
    else if (l16 >= 8 && l16 < 8 + CO) xr[(size_t)m * CO + (l16 - 8)] = v + br[l16 - 8];
  }
}

// -------------------- GATv2 edge phases --------------------

template <int CO>
__global__ void k_edge_scores(const int* __restrict__ src, const int* __restrict__ dst,
                              const float* __restrict__ edge_attr,
                              const float* __restrict__ loop_attr,
                              const float* __restrict__ xl, const float* __restrict__ xr,
                              const float* __restrict__ We, const float* __restrict__ att,
                              float* __restrict__ escore, float* __restrict__ emax) {
  int i = blockIdx.x * blockDim.x + threadIdx.x;
  if (i >= ET) return;
  int s, d; const float* ea;
  if (i < NE) { s = src[i]; d = dst[i]; ea = edge_attr + (size_t)i * FE; }
  else        { s = d = i - NE;        ea = loop_attr + (size_t)(i - NE) * FE; }
  float eav[FE];
#pragma unroll
  for (int j = 0; j < FE; ++j) eav[j] = ea[j];
  float score = 0.0f;
#pragma unroll
  for (int c = 0; c < CO; ++c) {
    float m = xl[(size_t)s * CO + c] + xr[(size_t)d * CO + c];
#pragma unroll
    for (int j = 0; j < FE; ++j) m = fmaf(eav[j], We[j * CO + c], m);
    float lm = (m > 0.0f) ? m : 0.2f * m;   // leaky_relu(0.2)
    score = fmaf(lm, att[c], score);
  }
  escore[i] = score;
  atomic_max_f32(&emax[d], score);
}

template <int CO>
__global__ void k_edge_softmax(const int* __restrict__ src, const int* __restrict__ dst,
                               const float* __restrict__ escore, const float* __restrict__ emax,
                               const float* __restrict__ xl,
                               float* __restrict__ denom, float* __restrict__ num) {
  int i = blockIdx.x * blockDim.x + threadIdx.x;
  if (i >= ET) return;
  int s, d;
  if (i < NE) { s = src[i]; d = dst[i]; } else { s = d = i - NE; }
  float ez = __expf(escore[i] - emax[d]);
  atomicAdd(&denom[d], ez);
#pragma unroll
  for (int c = 0; c < CO; ++c)
    atomicAdd(&num[(size_t)d * CO + c], ez * xl[(size_t)s * CO + c]);
}

template <int CO>
__global__ void k_node_finalize(const float* __restrict__ num, const float* __restrict__ denom,
                                const float* __restrict__ bias, float* __restrict__ hout) {
  int n = blockIdx.x * blockDim.x + threadIdx.x;
  if (n >= NN) return;
  float inv = 1.0f / denom[n];   // self-loop guarantees denom > 0
#pragma unroll
  for (int c = 0; c < CO; ++c)
    hout[(size_t)n * CO + c] = num[(size_t)n * CO + c] * inv + bias[c];
}

// -------------------- GraphNorm + ReLU (one block per graph, in place) -----

template <int CO>
__global__ void k_graph_norm(float* h, const float* __restrict__ w,
                             const float* __restrict__ b, const float* __restrict__ ms) {
  __shared__ float red[NG];
  const int g = blockIdx.x, t = threadIdx.x;
  const size_t row = (size_t)g * NG + t;
#pragma unroll
  for (int c = 0; c < CO; ++c) {
    float v = h[row * CO + c];
    red[t] = v; __syncthreads();
    for (int off = NG / 2; off > 0; off >>= 1) {
      if (t < off) red[t] += red[t + off];
      __syncthreads();
    }
    float mean = red[0] * (1.0f / NG);
    __syncthreads();
    float out = v - ms[c] * mean;
    red[t] = out * out; __syncthreads();
    for (int off = NG / 2; off > 0; off >>= 1) {
      if (t < off) red[t] += red[t + off];
      __syncthreads();
    }
    float var = red[0] * (1.0f / NG);
    __syncthreads();
    float y = w[c] * out * rsqrtf(var + 1e-5f) + b[c];
    h[row * CO + c] = fmaxf(y, 0.0f);
  }
}

// -------------------- dueling head: advantage MLP --------------------

__global__ void k_adv(const float* __restrict__ h1, const float* __restrict__ h2,
                      const float* __restrict__ h3,
                      const float* __restrict__ Wa1, const float* __restrict__ ba1,
                      const float* __restrict__ Wa2, const float* __restrict__ ba2,
                      float* __restrict__ adv) {
  int n = blockIdx.x * blockDim.x + threadIdx.x;
  if (n >= NN) return;
  float xc[18];
#pragma unroll
  for (int k = 0; k < 8; ++k) xc[k]      = h1[(size_t)n * 8 + k];
#pragma unroll
  for (int k = 0; k < 6; ++k) xc[8 + k]  = h2[(size_t)n * 6 + k];
#pragma unroll
  for (int k = 0; k < 4; ++k) xc[14 + k] = h3[(size_t)n * 4 + k];
  float a = ba2[0];
#pragma unroll
  for (int j = 0; j < 8; ++j) {
    float hj = ba1[j];
#pragma unroll
    for (int k = 0; k < 18; ++k) hj = fmaf(xc[k], Wa1[k * 8 + j], hj);
    a = fmaf(fmaxf(hj, 0.0f), Wa2[j], a);
  }
  adv[n] = a;
}

// -------------------- diff-pool branch --------------------
// adj @ xd == edge scatter: agg[src] += x3[dst], outdeg[src] += 1

__global__ void k_pool_scatter(const int* __restrict__ src, const int* __restrict__ dst,
                               const float* __restrict__ h3,
                               float* __restrict__ agg, float* __restrict__ outdeg) {
  int e = blockIdx.x * blockDim.x + threadIdx.x;
  if (e >= NE) return;
  int s = src[e], d = dst[e];
  atomicAdd(&outdeg[s], 1.0f);
#pragma unroll
  for (int f = 0; f < 4; ++f)
    atomicAdd(&agg[(size_t)s * 4 + f], h3[(size_t)d * 4 + f]);
}

__global__ void k_assign(const float* __restrict__ h3, const float* __restrict__ agg,
                         const float* __restrict__ outdeg,
                         const float* __restrict__ Wrel, const float* __restrict__ Wroot,
                         const float* __restrict__ broot, float* __restrict__ sass) {
  int n = blockIdx.x * blockDim.x + threadIdx.x;
  if (n >= NN) return;
  float inv = 1.0f / fmaxf(outdeg[n], 1.0f);
  float af[4], xf[4];
#pragma unroll
  for (int f = 0; f < 4; ++f) { af[f] = agg[(size_t)n * 4 + f] * inv; xf[f] = h3[(size_t)n * 4 + f]; }
  float logit[6], mx = -3.4e38f;
#pragma unroll
  for (int c = 0; c < 6; ++c) {
    float v = broot[c];
#pragma unroll
    for (int f = 0; f < 4; ++f) v = fmaf(af[f], Wrel[f * 6 + c], fmaf(xf[f], Wroot[f * 6 + c], v));
    logit[c] = v; mx = fmaxf(mx, v);
  }
  float e[6], sum = 0.0f;
#pragma unroll
  for (int c = 0; c < 6; ++c) { e[c] = __expf(logit[c] - mx); sum += e[c]; }
  float isum = 1.0f / sum;
#pragma unroll
  for (int c = 0; c < 6; ++c) sass[(size_t)n * 6 + c] = e[c] * isum;
}

// pooled = s^T @ xd per graph, value MLP, and mean-advantage -- one block/graph
__global__ void k_graph_pool(const float* __restrict__ sass, const float* __restrict__ h3,
                             const float* __restrict__ adv,
                             const float* __restrict__ Wv1, const float* __restrict__ bv1,
                             const float* __restrict__ Wv2, const float* __restrict__ bv2,
                             float* __restrict__ value, float* __restrict__ mean_adv) {
  __shared__ float red[NG];
  __shared__ float pooled[24];
  const int g = blockIdx.x, t = threadIdx.x;
  const size_t row = (size_t)g * NG + t;
  float sl[6], xl4[4];
#pragma unroll
  for (int c = 0; c < 6; ++c) sl[c] = sass[row * 6 + c];
#pragma unroll
  for (int f = 0; f < 4; ++f) xl4[f] = h3[row * 4 + f];
#pragma unroll
  for (int c = 0; c < 6; ++c) {
#pragma unroll
    for (int f = 0; f < 4; ++f) {
      red[t] = sl[c] * xl4[f]; __syncthreads();
      for (int off = NG / 2; off > 0; off >>= 1) {
        if (t < off) red[t] += red[t + off];
        __syncthreads();
      }
      if (t == 0) pooled[c * 4 + f] = red[0];
      __syncthreads();
    }
  }
  red[t] = adv[row]; __syncthreads();
  for (int off = NG / 2; off > 0; off >>= 1) {
    if (t < off) red[t] += red[t + off];
    __syncthreads();
  }
  if (t == 0) {
    mean_adv[g] = red[0] * (1.0f / NG);
    float pv[4];
#pragma unroll
    for (int f = 0; f < 4; ++f) {
      float acc = 0.0f;
#pragma unroll
      for (int c = 0; c < 6; ++c) acc += pooled[c * 4 + f];
      pv[f] = acc * (1.0f / 6.0f);
    }
    float val = bv2[0];
#pragma unroll
    for (int j = 0; j < 8; ++j) {
      float hj = bv1[j];
#pragma unroll
      for (int f = 0; f < 4; ++f) hj = fmaf(pv[f], Wv1[f * 8 + j], hj);
      val = fmaf(fmaxf(hj, 0.0f), Wv2[j], val);
    }
    value[g] = val;
  }
}

__global__ void k_output(const float* __restrict__ value, const float* __restrict__ adv,
                         const float* __restrict__ mean_adv, float* __restrict__ out) {
  int n = blockIdx.x * blockDim.x + threadIdx.x;
  if (n >= NN) return;
  int g = n >> 9;   // NG == 512
  out[n] = value[g] + adv[n] - mean_adv[g];
}

// -------------------- host launch --------------------

static inline int cdiv(int a, int b) { return (a + b - 1) / b; }

// params flat element counts, in setup_inputs insertion order
static const int PSIZES[41] = {
  128, 8, 128, 8, 64, 8, 8, 8, 8, 8,    // layer1: Wl bl Wr br We att bias gnw gnb gnm
  48, 6, 48, 6, 48, 6, 6, 6, 6, 6,      // layer2
  24, 4, 24, 4, 32, 4, 4, 4, 4, 4,      // layer3
  24, 24, 6, 32, 8, 8, 1, 144, 8, 8, 1  // Wrel Wroot broot Wv1 bv1 Wv2 bv2 Wa1 ba1 Wa2 ba2
};

extern "C" void kernel_launch(void* const* d_in, const int* in_sizes, int n_in,
                              void* d_out, int out_size, void* d_ws, size_t ws_size,
                              hipStream_t stream) {
  const float* x         = (const float*)d_in[0];
  const int*   ei        = (const int*)d_in[1];
  const int*   src       = ei;
  const int*   dst       = ei + NE;
  const float* edge_attr = (const float*)d_in[2];
  // d_in[3] = batch: graphs are contiguous (node n -> graph n>>9), not needed.

  // Resolve param pointers: either one d_in entry per tensor, or one blob.
  const float* P[41];
  if (n_in >= 45) {
    for (int i = 0; i < 41; ++i) P[i] = (const float*)d_in[4 + i];
  } else {
    const float* blob = (const float*)d_in[4];
    size_t o = 0;
    for (int i = 0; i < 41; ++i) { P[i] = blob + o; o += (size_t)PSIZES[i]; }
  }
  const float *Wrel = P[30], *Wroot = P[31], *broot = P[32];
  const float *Wv1 = P[33], *bv1 = P[34], *Wv2 = P[35], *bv2 = P[36];
  const float *Wa1 = P[37], *ba1 = P[38], *Wa2 = P[39], *ba2 = P[40];

  // workspace (floats), bump-allocated; accumulators re-zeroed every call
  float* w = (float*)d_ws;
  size_t off = 0;
  auto alloc = [&](size_t n) { float* p = w + off; off += n; return p; };
  float* deg       = alloc(NN);
  float* loop_attr = alloc((size_t)NN * FE);   // contiguous after deg
  float* xl        = alloc((size_t)NN * 8);
  float* xr        = alloc((size_t)NN * 8);
  float* escore    = alloc((size_t)ET);
  float* emax      = alloc(NN);
  float* denom     = alloc(NN);                // contiguous with num
  float* num       = alloc((size_t)NN * 8);
  float* h1        = alloc((size_t)NN * 8);
  float* h2        = alloc((size_t)NN * 6);
  float* h3        = alloc((size_t)NN * 4);
  float* advv      = alloc(NN);
  float* agg       = alloc((size_t)NN * 4);    // contiguous with outdeg
  float* outdeg    = alloc(NN);
  float* sass      = alloc((size_t)NN * 6);
  float* valueb    = alloc(B_GRAPHS);
  float* madv      = alloc(B_GRAPHS);
  (void)ws_size; (void)in_sizes; (void)out_size;

  const int TB = 256;
  const float NEG_INF = -__builtin_inff();

  // ---- self-loop attrs (layer independent) ----
  k_fill<<<cdiv(NN * 9, TB), TB, 0, stream>>>(deg, 0.0f, NN * 9); // deg + loop_attr
  k_loop_scatter<<<cdiv(NE, TB), TB, 0, stream>>>(dst, edge_attr, deg, loop_attr);
  k_loop_norm<<<cdiv(NN, TB), TB, 0, stream>>>(loop_attr, deg);

  const int WAVE_BLOCKS = NN / 16 / (TB / 32);   // 1024: one wave per 16-node tile
  const int EB = cdiv(ET, TB);

  // ---- layer 1: 16 -> 8 ----
  k_node_transform<16, 8><<<WAVE_BLOCKS, TB, 0, stream>>>(x, P[0], P[1], P[2], P[3], xl, xr);
  k_fill<<<cdiv(NN, TB), TB, 0, stream>>>(emax, NEG_INF, NN);
  k_fill<<<cdiv(NN * 9, TB), TB, 0, stream>>>(denom, 0.0f, NN * 9); // denom + num
  k_edge_scores<8><<<EB, TB, 0, stream>>>(src, dst, edge_attr, loop_attr, xl, xr, P[4], P[5], escore, emax);
  k_edge_softmax<8><<<EB, TB, 0, stream>>>(src, dst, escore, emax, xl, denom, num);
  k_node_finalize<8><<<cdiv(NN, TB), TB, 0, stream>>>(num, denom, P[6], h1);
  k_graph_norm<8><<<B_GRAPHS, NG, 0, stream>>>(h1, P[7], P[8], P[9]);

  // ---- layer 2: 8 -> 6 ----
  k_node_transform<8, 6><<<WAVE_BLOCKS, TB, 0, stream>>>(h1, P[10], P[11], P[12], P[13], xl, xr);
  k_fill<<<cdiv(NN, TB), TB, 0, stream>>>(emax, NEG_INF, NN);
  k_fill<<<cdiv(NN * 9, TB), TB, 0, stream>>>(denom, 0.0f, NN * 9);
  k_edge_scores<6><<<EB, TB, 0, stream>>>(src, dst, edge_attr, loop_attr, xl, xr, P[14], P[15], escore, emax);
  k_edge_softmax<6><<<EB, TB, 0, stream>>>(src, dst, escore, emax, xl, denom, num);
  k_node_finalize<6><<<cdiv(NN, TB), TB, 0, stream>>>(num, denom, P[16], h2);
  k_graph_norm<6><<<B_GRAPHS, NG, 0, stream>>>(h2, P[17], P[18], P[19]);

  // ---- layer 3: 6 -> 4 ----
  k_node_transform<6, 4><<<WAVE_BLOCKS, TB, 0, stream>>>(h2, P[20], P[21], P[22], P[23], xl, xr);
  k_fill<<<cdiv(NN, TB), TB, 0, stream>>>(emax, NEG_INF, NN);
  k_fill<<<cdiv(NN * 9, TB), TB, 0, stream>>>(denom, 0.0f, NN * 9);
  k_edge_scores<4><<<EB, TB, 0, stream>>>(src, dst, edge_attr, loop_attr, xl, xr, P[24], P[25], escore, emax);
  k_edge_softmax<4><<<EB, TB, 0, stream>>>(src, dst, escore, emax, xl, denom, num);
  k_node_finalize<4><<<cdiv(NN, TB), TB, 0, stream>>>(num, denom, P[26], h3);
  k_graph_norm<4><<<B_GRAPHS, NG, 0, stream>>>(h3, P[27], P[28], P[29]);

  // ---- head ----
  k_adv<<<cdiv(NN, TB), TB, 0, stream>>>(h1, h2, h3, Wa1, ba1, Wa2, ba2, advv);
  k_fill<<<cdiv(NN * 5, TB), TB, 0, stream>>>(agg, 0.0f, NN * 5);   // agg + outdeg
  k_pool_scatter<<<cdiv(NE, TB), TB, 0, stream>>>(src, dst, h3, agg, outdeg);
  k_assign<<<cdiv(NN, TB), TB, 0, stream>>>(h3, agg, outdeg, Wrel, Wroot, broot, sass);
  k_graph_pool<<<B_GRAPHS, NG, 0, stream>>>(sass, h3, advv, Wv1, bv1, Wv2, bv2, valueb, madv);
  k_output<<<cdiv(NN, TB), TB, 0, stream>>>(valueb, advv, madv, (float*)d_out);
}